// VectorQuantizer_22978075034375
// MI455X (gfx1250) — compile-verified
//
#include <hip/hip_runtime.h>
#include <hip/hip_bf16.h>

typedef float v2f __attribute__((ext_vector_type(2)));
typedef float v8f __attribute__((ext_vector_type(8)));

#define N_TOK 65536
#define K_CODES 1024
#define D_DIM 64
#define DECAY 0.99f
#define BETA 0.25f

// d_out float offsets
#define ZQ_OFF   0
#define LOSS_OFF 4194304
#define IDX_OFF  4194305
#define CNT_OFF  4259841
#define AVG_OFF  4260865

// workspace float offsets
#define WS_WSQ   0
#define WS_CNT   1024
#define WS_LOSS  2048
#define WS_AVG   4096
#define WS_IDX   (4096 + 65536)            // int region
#define WS_TOTAL_FLOATS (4096 + 65536 + 65536)

// ---------------------------------------------------------------------------
// Kernel 0: wsq[k] = ||W[k]||^2.  One wave per code; lane handles 2 dims.
// ---------------------------------------------------------------------------
__global__ __launch_bounds__(256) void vq_wsq_kernel(const float* __restrict__ emb,
                                                     float* __restrict__ wsq) {
    int lane = threadIdx.x & 31;
    int wave = threadIdx.x >> 5;
    int code = blockIdx.x * 8 + wave;
    v2f e = *(const v2f*)(emb + code * D_DIM + lane * 2);
    float s = e.x * e.x + e.y * e.y;
    #pragma unroll
    for (int off = 16; off >= 1; off >>= 1) s += __shfl_xor(s, off, 32);
    if (lane == 0) wsq[code] = s;
}

// ---------------------------------------------------------------------------
// Kernel 1: WMMA distance + argmin. One wave owns 32 tokens (two 16-row
// tiles sharing B fragments -> 32 WMMAs per 17 VMEM ops) vs all 1024 codes.
// S = z_e @ W^T via V_WMMA_F32_16X16X4_F32; track min(wsq - 2S) per lane.
// ---------------------------------------------------------------------------
__global__ __launch_bounds__(256) void vq_argmin_kernel(const float* __restrict__ ze,
                                                        const float* __restrict__ emb,
                                                        const float* __restrict__ wsq,
                                                        int* __restrict__ idx_ws,
                                                        float* __restrict__ idx_out) {
    int lane = threadIdx.x & 31;
    int wave = threadIdx.x >> 5;
    int tokBase = (blockIdx.x * 8 + wave) * 32;      // 32 tokens per wave
    int m    = lane & 15;   // row (A) / column-code (B) within tile
    int half = lane >> 4;   // selects K pair {0,1} vs {2,3} within each 4-chunk

    // A fragments for two token tiles: 16 chunks of K=4, float2/lane/chunk.
    v2f a0[16], a1[16];
    const float* z0 = ze + (size_t)(tokBase + m) * D_DIM + half * 2;
    const float* z1 = ze + (size_t)(tokBase + 16 + m) * D_DIM + half * 2;
    #pragma unroll
    for (int c = 0; c < 16; ++c) { a0[c] = *(const v2f*)(z0 + c * 4);
                                   a1[c] = *(const v2f*)(z1 + c * 4); }

    float best0[8], best1[8];
    int   bidx0[8], bidx1[8];
    #pragma unroll
    for (int r = 0; r < 8; ++r) {
        best0[r] = 3.4e38f; bidx0[r] = 0;
        best1[r] = 3.4e38f; bidx1[r] = 0;
    }

    for (int t = 0; t < 64; ++t) {                   // 64 tiles of 16 codes
        const float* brow = emb + (size_t)(t * 16 + m) * D_DIM + half * 2;
        v2f b[16];
        #pragma unroll
        for (int c = 0; c < 16; ++c) b[c] = *(const v2f*)(brow + c * 4);
        float wq = wsq[t * 16 + m];                  // ||W[code]||^2, code = t*16 + lane%16

        v8f acc0 = {}, acc1 = {};
        #pragma unroll
        for (int c = 0; c < 16; ++c) {               // two independent chains
            acc0 = __builtin_amdgcn_wmma_f32_16x16x4_f32(false, a0[c], false, b[c],
                                                         (short)0, acc0, false, false);
            acc1 = __builtin_amdgcn_wmma_f32_16x16x4_f32(false, a1[c], false, b[c],
                                                         (short)0, acc1, false, false);
        }

        int code = t * 16 + m;
        #pragma unroll
        for (int r = 0; r < 8; ++r) {
            float v0 = fmaf(-2.0f, acc0[r], wq);     // ||z||^2 constant per row: omit
            bool l0 = v0 < best0[r];
            best0[r] = l0 ? v0   : best0[r];
            bidx0[r] = l0 ? code : bidx0[r];
            float v1 = fmaf(-2.0f, acc1[r], wq);
            bool l1 = v1 < best1[r];
            best1[r] = l1 ? v1   : best1[r];
            bidx1[r] = l1 ? code : bidx1[r];
        }
    }

    // Reduce (min,argmin) across the 16 lanes of each half-wave.
    #pragma unroll
    for (int off = 1; off < 16; off <<= 1) {
        #pragma unroll
        for (int r = 0; r < 8; ++r) {
            float ov = __shfl_xor(best0[r], off, 32);
            int   oi = __shfl_xor(bidx0[r], off, 32);
            bool take = (ov < best0[r]) || (ov == best0[r] && oi < bidx0[r]);
            best0[r] = take ? ov : best0[r];
            bidx0[r] = take ? oi : bidx0[r];
            float pv = __shfl_xor(best1[r], off, 32);
            int   pi = __shfl_xor(bidx1[r], off, 32);
            bool tk1 = (pv < best1[r]) || (pv == best1[r] && pi < bidx1[r]);
            best1[r] = tk1 ? pv : best1[r];
            bidx1[r] = tk1 ? pi : bidx1[r];
        }
    }

    if ((lane & 15) == 0) {                          // lane 0 -> rows 0..7, lane 16 -> 8..15
        int rowBase = tokBase + half * 8;
        #pragma unroll
        for (int r = 0; r < 8; ++r) {
            idx_ws[rowBase + r]       = bidx0[r];
            idx_out[rowBase + r]      = (float)bidx0[r];
            idx_ws[rowBase + 16 + r]  = bidx1[r];
            idx_out[rowBase + 16 + r] = (float)bidx1[r];
        }
    }
}

// ---------------------------------------------------------------------------
// Kernel 2: gather z_q, commitment partial sum, scatter-add count/avg.
// One wave per token; lane handles 2 dims (float2).
// ---------------------------------------------------------------------------
__global__ __launch_bounds__(256) void vq_gather_kernel(const float* __restrict__ ze,
                                                        const float* __restrict__ emb,
                                                        const int* __restrict__ idx_ws,
                                                        float* __restrict__ zq_out,
                                                        float* __restrict__ cnt_ws,
                                                        float* __restrict__ avg_ws,
                                                        float* __restrict__ loss_ws) {
    int lane = threadIdx.x & 31;
    int wave = threadIdx.x >> 5;
    int token = blockIdx.x * 8 + wave;
    int code = idx_ws[token];

    v2f q = *(const v2f*)(emb + (size_t)code * D_DIM + lane * 2);
    v2f z = *(const v2f*)(ze + (size_t)token * D_DIM + lane * 2);
    *(v2f*)(zq_out + (size_t)token * D_DIM + lane * 2) = q;

    float dx = z.x - q.x, dy = z.y - q.y;
    float s = dx * dx + dy * dy;
    #pragma unroll
    for (int off = 16; off >= 1; off >>= 1) s += __shfl_xor(s, off, 32);

    atomicAdd(avg_ws + (size_t)code * D_DIM + lane * 2,     z.x);
    atomicAdd(avg_ws + (size_t)code * D_DIM + lane * 2 + 1, z.y);
    if (lane == 0) {
        atomicAdd(loss_ws, s);
        atomicAdd(cnt_ws + code, 1.0f);
    }
}

// ---------------------------------------------------------------------------
// Kernel 3: EMA blends + loss finalize.
// ---------------------------------------------------------------------------
__global__ __launch_bounds__(256) void vq_ema_kernel(const float* __restrict__ ema_count,
                                                     const float* __restrict__ ema_avg,
                                                     const float* __restrict__ cnt_ws,
                                                     const float* __restrict__ avg_ws,
                                                     const float* __restrict__ loss_ws,
                                                     float* __restrict__ out) {
    int i = blockIdx.x * 256 + threadIdx.x;
    if (i == 0)
        out[LOSS_OFF] = BETA * loss_ws[0] / ((float)N_TOK * (float)D_DIM);
    if (i < K_CODES)
        out[CNT_OFF + i] = ema_count[i] * DECAY + (1.0f - DECAY) * cnt_ws[i];
    if (i < K_CODES * D_DIM)
        out[AVG_OFF + i] = ema_avg[i] * DECAY + (1.0f - DECAY) * avg_ws[i];
}

// ---------------------------------------------------------------------------
extern "C" void kernel_launch(void* const* d_in, const int* in_sizes, int n_in,
                              void* d_out, int out_size, void* d_ws, size_t ws_size,
                              hipStream_t stream) {
    const float* ze        = (const float*)d_in[0];
    const float* emb       = (const float*)d_in[1];
    const float* ema_count = (const float*)d_in[2];
    const float* ema_avg   = (const float*)d_in[3];
    float* out = (float*)d_out;
    float* ws  = (float*)d_ws;

    float* wsq  = ws + WS_WSQ;
    float* cnt  = ws + WS_CNT;
    float* loss = ws + WS_LOSS;
    float* avg  = ws + WS_AVG;
    int*   idxw = (int*)(ws + WS_IDX);

    hipMemsetAsync(d_ws, 0, (size_t)WS_TOTAL_FLOATS * sizeof(float), stream);

    vq_wsq_kernel<<<K_CODES / 8, 256, 0, stream>>>(emb, wsq);
    vq_argmin_kernel<<<N_TOK / 256, 256, 0, stream>>>(ze, emb, wsq, idxw, out + IDX_OFF);
    vq_gather_kernel<<<N_TOK / 8, 256, 0, stream>>>(ze, emb, idxw, out + ZQ_OFF,
                                                    cnt, avg, loss);
    vq_ema_kernel<<<(K_CODES * D_DIM + 255) / 256, 256, 0, stream>>>(
        ema_count, ema_avg, cnt, avg, loss, out);
}